// S4optimizer_14276471292495
// MI455X (gfx1250) — compile-verified
//
#include <hip/hip_runtime.h>
#include <math.h>

// ---------------------------------------------------------------------------
// S4D model on MI455X (gfx1250), fp32 end-to-end, V_WMMA_F32_16X16X4_F32.
//   embed GEMM -> [4-lane-split SSM scan -> fused outW-GEMM+GLU+LN] x4 -> proj GEMM
// A-strip staging uses GLOBAL_LOAD_ASYNC_TO_LDS_B128 (ASYNCcnt) when available.
// ---------------------------------------------------------------------------

typedef __attribute__((ext_vector_type(2))) float v2f;
typedef __attribute__((ext_vector_type(8))) float v8f;
typedef __attribute__((ext_vector_type(4))) int   v4i;

#define S_   2048
#define T_   32
#define H_   256
#define N2_  32
#define NL_  4
#define Z_   128
#define L_   2049            // S + 1
#define TH_  (T_ * H_)       // 8192
#define LT_  (L_ * T_)       // 65568 (divisible by 16)
#define AS_  260             // LDS stride (floats) for staged A strip: bank-conflict-free b64 frags
#define HS_  516             // LDS stride (floats) for 16x512 result tile: conflict-free stores

static __device__ __forceinline__ v2f ld2(const float* p) { return *(const v2f*)p; }

// --- CDNA5 async global->LDS staging (bypasses VGPRs, tracked by ASYNCcnt) --
#if defined(__AMDGCN__) && defined(__has_builtin)
#if __has_builtin(__builtin_amdgcn_global_load_async_to_lds_b128)
#define ASYNC_LDS 1
#endif
#endif

#if ASYNC_LDS
typedef __attribute__((address_space(1))) v4i gv4i_t;   // global (prints as __device__)
typedef __attribute__((address_space(3))) v4i lv4i_t;   // LDS    (prints as __shared__)
static __device__ __forceinline__ void async_cp16(const float* g, float* l) {
    __builtin_amdgcn_global_load_async_to_lds_b128((gv4i_t*)g, (lv4i_t*)l, 0, 0);
}
static __device__ __forceinline__ void async_wait0() {
    asm volatile("s_wait_asynccnt 0" ::: "memory");
}
#endif

// --- u[0, t, h] = x0[h] -----------------------------------------------------
__global__ void fill_u0(const float* __restrict__ x0, float* __restrict__ u) {
    int tid = blockIdx.x * blockDim.x + threadIdx.x;
    if (tid < TH_) u[tid] = x0[tid & (H_ - 1)];
}

// --- embed: u[(s+1)*T+t, h] = concat(x,y)[s,t,:] @ emb_W + emb_b ------------
__global__ void embed_kernel(const float* __restrict__ x, const float* __restrict__ y,
                             const float* __restrict__ W, const float* __restrict__ b,
                             float* __restrict__ u) {
    const int lane = threadIdx.x;
    const int row0 = blockIdx.x * 16;         // over S*T
    const int col0 = blockIdx.y * 16;         // over H
    const int half = lane >> 4;
    const int mn   = lane & 15;
    v8f c = {0.f, 0.f, 0.f, 0.f, 0.f, 0.f, 0.f, 0.f};
    const int gA = row0 + mn;
    #pragma unroll
    for (int kk = 0; kk < 64; kk += 4) {
        const int k0 = kk + half * 2;
        const float* asrc = (k0 < 32) ? (x + (size_t)gA * 32 + k0)
                                      : (y + (size_t)gA * 32 + (k0 - 32));
        v2f a = ld2(asrc);
        v2f bf;
        bf.x = W[(size_t)k0 * H_ + col0 + mn];
        bf.y = W[(size_t)(k0 + 1) * H_ + col0 + mn];
        c = __builtin_amdgcn_wmma_f32_16x16x4_f32(false, a, false, bf, (short)0, c, false, false);
    }
    const float bias = b[col0 + mn];
    #pragma unroll
    for (int r = 0; r < 8; ++r) {
        const int g = row0 + r + 8 * half;    // g = s*T + t ; stored at l = s+1 -> row g+T
        u[(size_t)(g + T_) * H_ + col0 + mn] = c[r] + bias;
    }
}

// --- per-layer S4D parameters: lambda = exp(dt*A), Ct = 2*C*(lambda-1)/A ----
__global__ void s4_params(const float* __restrict__ log_dt, const float* __restrict__ log_A_real,
                          const float* __restrict__ A_imag, const float* __restrict__ C_re,
                          const float* __restrict__ C_im,
                          float* __restrict__ lre, float* __restrict__ lim,
                          float* __restrict__ cre, float* __restrict__ cim) {
    int tid = blockIdx.x * blockDim.x + threadIdx.x;
    if (tid >= NL_ * H_ * N2_) return;
    const int hn = tid % (H_ * N2_);
    const int i  = tid / (H_ * N2_);
    const int h  = hn / N2_;
    const float dt  = expf(log_dt[i * H_ + h]);
    const float Are = -expf(log_A_real[tid]);
    const float Aim = A_imag[tid];
    const float zr = dt * Are, zi = dt * Aim;
    const float em = expf(zr);
    const float Lre = em * cosf(zi);
    const float Lim = em * sinf(zi);
    const float dr = Lre - 1.0f, di = Lim;
    const float den = Are * Are + Aim * Aim;
    const float qr = (dr * Are + di * Aim) / den;
    const float qi = (di * Are - dr * Aim) / den;
    const float Cr = C_re[tid], Ci = C_im[tid];
    lre[tid] = Lre;  lim[tid] = Lim;
    cre[tid] = 2.0f * (Cr * qr - Ci * qi);
    cim[tid] = 2.0f * (Cr * qi + Ci * qr);
}

// --- diagonal SSM scan + D-skip + exact GELU --------------------------------
// 4 lanes per (t,h): 8 complex modes each; cross-lane combine via shfl_xor.
// 4*TH_ = 32768 threads -> 1024 waves on the sequential critical path.
__global__ void s4_scan(const float* __restrict__ u, float* __restrict__ hb,
                        const float* __restrict__ lre, const float* __restrict__ lim,
                        const float* __restrict__ cre, const float* __restrict__ cim,
                        const float* __restrict__ Dv) {
    const int gid    = blockIdx.x * blockDim.x + threadIdx.x;  // 0 .. 4*TH_-1
    const int sub    = gid & 3;                                // mode group 0..3
    const int pairid = gid >> 2;                               // t*H + h
    const int h      = pairid & (H_ - 1);
    float Lr[8], Li[8], Cr[8], Ci[8], sr[8], si[8];
    const int pb = h * N2_ + sub * 8;
    #pragma unroll
    for (int n = 0; n < 8; ++n) {
        Lr[n] = lre[pb + n]; Li[n] = lim[pb + n];
        Cr[n] = cre[pb + n]; Ci[n] = cim[pb + n];
        sr[n] = 0.f; si[n] = 0.f;
    }
    const float Dh = Dv[h];
    size_t idx = (size_t)pairid;
    for (int l = 0; l < L_; ++l) {
        const float uv = u[idx];
        float yv = 0.f;
        #pragma unroll
        for (int n = 0; n < 8; ++n) {
            const float nr = __builtin_fmaf(Lr[n], sr[n], -Li[n] * si[n]) + uv;
            const float ni = __builtin_fmaf(Lr[n], si[n],  Li[n] * sr[n]);
            sr[n] = nr; si[n] = ni;
            yv = __builtin_fmaf(Cr[n], nr, yv);
            yv = __builtin_fmaf(-Ci[n], ni, yv);
        }
        yv += __shfl_xor(yv, 1, 32);
        yv += __shfl_xor(yv, 2, 32);
        const float hv = __builtin_fmaf(Dh, uv, yv);
        if (sub == 0)
            hb[idx] = 0.5f * hv * (1.0f + erff(hv * 0.70710678118654752f));  // exact GELU
        idx += TH_;
    }
}

// --- fused: h2 = h @ out_W^T + out_b ; GLU ; +residual ; channel LayerNorm --
// One block = 16 rows x all 512 outputs. A strip async-staged into LDS, shared
// by 8 waves; each wave: 16x64 strip, one A fragment ds_load reused across 4
// WMMAs. Result tile stays in LDS; 16-thread-per-row epilogue does GLU+LN and
// updates u in place (no h2 round-trip through memory).
__global__ __launch_bounds__(256) void out_glu_ln(const float* __restrict__ hbuf,
                                                  const float* __restrict__ W,
                                                  const float* __restrict__ bias,
                                                  const float* __restrict__ gam,
                                                  const float* __restrict__ bet,
                                                  float* __restrict__ u) {
    __shared__ float At[16 * AS_];   // 16 x 256 A strip, padded
    __shared__ float Ht[16 * HS_];   // 16 x 512 result tile, padded
    const int tid  = threadIdx.x;
    const int row0 = blockIdx.x * 16;

    {   // stage A strip (coalesced: 16 contiguous floats per thread)
        const int r  = tid >> 4;
        const int c0 = (tid & 15) * 16;
        const float* src = hbuf + (size_t)(row0 + r) * H_ + c0;
        float* dst = At + r * AS_ + c0;
#if ASYNC_LDS
        #pragma unroll
        for (int i = 0; i < 16; i += 4)
            async_cp16(src + i, dst + i);       // global_load_async_to_lds_b128
        async_wait0();                          // s_wait_asynccnt 0
#else
        #pragma unroll
        for (int i = 0; i < 16; i += 4)
            *(float4*)(dst + i) = *(const float4*)(src + i);
#endif
    }
    __syncthreads();

    const int wave = tid >> 5, lane = tid & 31;
    const int half = lane >> 4, mn = lane & 15;
    v8f acc[4];
    #pragma unroll
    for (int j = 0; j < 4; ++j) acc[j] = (v8f){0.f,0.f,0.f,0.f,0.f,0.f,0.f,0.f};
    const float* wr0 = W + (size_t)(wave * 64 +  0 + mn) * H_;   // B^T columns (rows of out_W)
    const float* wr1 = W + (size_t)(wave * 64 + 16 + mn) * H_;
    const float* wr2 = W + (size_t)(wave * 64 + 32 + mn) * H_;
    const float* wr3 = W + (size_t)(wave * 64 + 48 + mn) * H_;
    const float* arow = At + mn * AS_;
    __builtin_prefetch(wr0, 0, 1);
    #pragma unroll 4
    for (int kk = 0; kk < H_; kk += 4) {
        const int k0 = kk + half * 2;
        v2f a = *(const v2f*)(arow + k0);            // ds_load_b64, reused x4
        acc[0] = __builtin_amdgcn_wmma_f32_16x16x4_f32(false, a, false, ld2(wr0 + k0), (short)0, acc[0], false, false);
        acc[1] = __builtin_amdgcn_wmma_f32_16x16x4_f32(false, a, false, ld2(wr1 + k0), (short)0, acc[1], false, false);
        acc[2] = __builtin_amdgcn_wmma_f32_16x16x4_f32(false, a, false, ld2(wr2 + k0), (short)0, acc[2], false, false);
        acc[3] = __builtin_amdgcn_wmma_f32_16x16x4_f32(false, a, false, ld2(wr3 + k0), (short)0, acc[3], false, false);
    }
    #pragma unroll
    for (int j = 0; j < 4; ++j) {
        const int col = wave * 64 + j * 16 + mn;
        const float bs = bias[col];
        #pragma unroll
        for (int r = 0; r < 8; ++r)
            Ht[(r + 8 * half) * HS_ + col] = acc[j][r] + bs;
    }
    __syncthreads();

    {   // GLU + residual + LayerNorm over H=256; 16 threads per row
        const int r  = tid >> 4;
        const int c0 = (tid & 15) * 16;
        const float* urow = u + (size_t)(row0 + r) * H_;
        float v[16];
        float s = 0.f, s2 = 0.f;
        #pragma unroll
        for (int i = 0; i < 16; ++i) {
            const int ch = c0 + i;
            const float a  = Ht[r * HS_ + ch];
            const float g  = Ht[r * HS_ + H_ + ch];
            const float vv = a * (1.0f / (1.0f + expf(-g))) + urow[ch];
            v[i] = vv; s += vv; s2 = __builtin_fmaf(vv, vv, s2);
        }
        #pragma unroll
        for (int o = 1; o < 16; o <<= 1) {
            s  += __shfl_xor(s,  o, 16);
            s2 += __shfl_xor(s2, o, 16);
        }
        const float mean = s * (1.0f / H_);
        const float var  = s2 * (1.0f / H_) - mean * mean;
        const float inv  = rsqrtf(var + 1e-5f);
        float* uout = u + (size_t)(row0 + r) * H_;
        #pragma unroll
        for (int i = 0; i < 16; ++i) {
            const int ch = c0 + i;
            uout[ch] = (v[i] - mean) * inv * gam[ch] + bet[ch];
        }
    }
}

// --- out = u @ proj_W + proj_b : 16-row strip per block, A async-staged -----
__global__ __launch_bounds__(256) void proj_kernel(const float* __restrict__ u,
                                                   const float* __restrict__ W,
                                                   const float* __restrict__ b,
                                                   float* __restrict__ out) {
    __shared__ float At[16 * AS_];
    const int tid  = threadIdx.x;
    const int row0 = blockIdx.x * 16;
    {   // stage A strip
        const int r  = tid >> 4;
        const int c0 = (tid & 15) * 16;
        const float* src = u + (size_t)(row0 + r) * H_ + c0;
        float* dst = At + r * AS_ + c0;
#if ASYNC_LDS
        #pragma unroll
        for (int i = 0; i < 16; i += 4)
            async_cp16(src + i, dst + i);
        async_wait0();
#else
        #pragma unroll
        for (int i = 0; i < 16; i += 4)
            *(float4*)(dst + i) = *(const float4*)(src + i);
#endif
    }
    __syncthreads();
    const int wave = tid >> 5, lane = tid & 31;
    const int half = lane >> 4, mn = lane & 15;
    const int col0 = wave * 16;                       // 8 waves cover Z=128
    v8f c = {0.f, 0.f, 0.f, 0.f, 0.f, 0.f, 0.f, 0.f};
    const float* arow = At + mn * AS_;
    #pragma unroll 4
    for (int kk = 0; kk < H_; kk += 4) {
        const int k0 = kk + half * 2;
        v2f a = *(const v2f*)(arow + k0);
        v2f bf;
        bf.x = W[(size_t)k0 * Z_ + col0 + mn];
        bf.y = W[(size_t)(k0 + 1) * Z_ + col0 + mn];
        c = __builtin_amdgcn_wmma_f32_16x16x4_f32(false, a, false, bf, (short)0, c, false, false);
    }
    const float bs = b[col0 + mn];
    #pragma unroll
    for (int r = 0; r < 8; ++r)
        out[(size_t)(row0 + r + 8 * half) * Z_ + col0 + mn] = c[r] + bs;
}

// ---------------------------------------------------------------------------
extern "C" void kernel_launch(void* const* d_in, const int* in_sizes, int n_in,
                              void* d_out, int out_size, void* d_ws, size_t ws_size,
                              hipStream_t stream) {
    const float* x          = (const float*)d_in[0];
    const float* y          = (const float*)d_in[1];
    const float* emb_W      = (const float*)d_in[2];
    const float* emb_b      = (const float*)d_in[3];
    const float* x0         = (const float*)d_in[4];
    const float* log_dt     = (const float*)d_in[5];
    const float* log_A_real = (const float*)d_in[6];
    const float* A_imag     = (const float*)d_in[7];
    const float* C_re       = (const float*)d_in[8];
    const float* C_im       = (const float*)d_in[9];
    const float* Dv         = (const float*)d_in[10];
    const float* out_W      = (const float*)d_in[11];
    const float* out_b      = (const float*)d_in[12];
    const float* ln_g       = (const float*)d_in[13];
    const float* ln_b       = (const float*)d_in[14];
    const float* proj_W     = (const float*)d_in[15];
    const float* proj_b     = (const float*)d_in[16];

    // workspace (fp32): u (L,T,H) | hb (L,T,H) | 4 param arrays  (~135 MB)
    float* u   = (float*)d_ws;
    float* hb  = u  + (size_t)TH_ * L_;
    float* lre = hb + (size_t)TH_ * L_;
    float* lim = lre + (size_t)NL_ * H_ * N2_;
    float* cre = lim + (size_t)NL_ * H_ * N2_;
    float* cim = cre + (size_t)NL_ * H_ * N2_;
    float* out = (float*)d_out;

    hipLaunchKernelGGL(fill_u0, dim3(TH_ / 256), dim3(256), 0, stream, x0, u);
    hipLaunchKernelGGL(embed_kernel, dim3((S_ * T_) / 16, H_ / 16), dim3(32), 0, stream,
                       x, y, emb_W, emb_b, u);
    hipLaunchKernelGGL(s4_params, dim3((NL_ * H_ * N2_) / 256), dim3(256), 0, stream,
                       log_dt, log_A_real, A_imag, C_re, C_im, lre, lim, cre, cim);

    for (int i = 0; i < NL_; ++i) {
        const int po = i * H_ * N2_;
        hipLaunchKernelGGL(s4_scan, dim3((4 * TH_) / 256), dim3(256), 0, stream,
                           u, hb, lre + po, lim + po, cre + po, cim + po, Dv + i * H_);
        hipLaunchKernelGGL(out_glu_ln, dim3(LT_ / 16), dim3(256), 0, stream,
                           hb, out_W + (size_t)i * 2 * H_ * H_, out_b + i * 2 * H_,
                           ln_g + i * H_, ln_b + i * H_, u);
    }
    hipLaunchKernelGGL(proj_kernel, dim3(LT_ / 16), dim3(256), 0, stream,
                       u, proj_W, proj_b, out);
}